// ConvCapsuleLayer_42545946035039
// MI455X (gfx1250) — compile-verified
//
#include <hip/hip_runtime.h>

typedef float v2f __attribute__((ext_vector_type(2)));
typedef float v8f __attribute__((ext_vector_type(8)));

// Problem constants (from reference):
//  x:[16,32,32,8,32]  W:[5,5,32,256]  b:[1,1,16,16]  out:[16,32,32,16,16]
#define HW      32
#define CIN     8
#define AI      32
#define NCH     256           // Co*Ao
#define KSTEPS  200           // K = 5*5*32 / 4
#define MTILE   16            // output positions per block

// ---------------------------------------------------------------------------
// Fully fused capsule conv + dynamic routing.
// Block = 256 threads = 8 wave32s. Each block owns 16 consecutive spatial
// positions (same b, same row) and:
//   Phase A: for each of the 8 input capsules, implicit-GEMM 5x5 conv via
//            V_WMMA_F32_16X16X4_F32 (each wave: 32 of the 256 channels),
//            accumulating the whole votes tile [8][16][256] in LDS (128 KB).
//   Phase B: 3 dynamic-routing iterations per position straight out of LDS.
// votes never touches HBM: traffic drops ~300MB -> ~35MB at 23.3 TB/s.
// ---------------------------------------------------------------------------
__global__ __launch_bounds__(256) void caps_fused_wmma(
    const float* __restrict__ x, const float* __restrict__ Wt,
    const float* __restrict__ bias, float* __restrict__ out)
{
    __shared__ float patch[5 * 20 * 32];          // [kh][col 0..19][ai], 12.8 KB
    __shared__ float votesLDS[CIN * MTILE * NCH]; // [ci][pos][ch], 128 KB
    __shared__ float logits[CIN][16];
    __shared__ float route[CIN][16];

    const int posBase = blockIdx.x * MTILE;       // global (b,h,w) flat base
    const int b   = posBase >> 10;                // / (32*32)
    const int hw0 = posBase & 1023;
    const int h   = hw0 >> 5;
    const int w0  = hw0 & 31;                     // 0 or 16

    const int t     = threadIdx.x;
    const int lane  = t & 31;
    const int wave  = t >> 5;                     // 0..7
    const int nbase = wave * 32;                  // this wave's 32 channels
    const int laneM = lane & 15;
    const int koff  = (lane >> 4) * 2;            // lanes 16-31 hold K+2/K+3
    const int mrow  = (lane >> 4) * 8;            // D rows: r / 8+r

    // ---------------- Phase A: conv -> votes (LDS) -------------------------
    for (int cin = 0; cin < CIN; ++cin) {
        __syncthreads();   // previous patch fully consumed
        // zero-padded patch: x[b, h+kh-2, w0+col-2, cin, ai]
        for (int i = t; i < 5 * 20 * 32; i += 256) {
            int kh  = i / 640;
            int rem = i - kh * 640;
            int col = rem >> 5;
            int ai  = rem & 31;
            int hin = h + kh - 2;
            int win = w0 + col - 2;
            float val = 0.0f;
            if (hin >= 0 && hin < HW && win >= 0 && win < HW)
                val = x[(((size_t)(b * HW + hin) * HW + win) * CIN + cin) * AI + ai];
            patch[i] = val;
        }
        __syncthreads();

        v8f acc0 = {0.f,0.f,0.f,0.f,0.f,0.f,0.f,0.f};
        v8f acc1 = {0.f,0.f,0.f,0.f,0.f,0.f,0.f,0.f};

        for (int step = 0; step < KSTEPS; ++step) {
            const int p25 = step >> 3;            // (kh,kw) index
            const int ai0 = (step & 7) * 4;       // ai base of this K-quad
            const int kh  = p25 / 5;
            const int kw  = p25 - kh * 5;

            // A 16x4 tile: row = laneM; VGPR0 = K+koff, VGPR1 = K+koff+1
            const float* ap = &patch[((kh * 20 + laneM + kw) * 32) + ai0 + koff];
            v2f a; a.x = ap[0]; a.y = ap[1];

            // B 4x16 tiles from W[k][ch] (819 KB, L2-resident)
            const int k0 = (kh * 5 + kw) * 32 + ai0;
            const float* bp = &Wt[(size_t)(k0 + koff) * NCH + nbase + laneM];
            v2f b0, b1;
            b0.x = bp[0];        b0.y = bp[NCH];
            b1.x = bp[16];       b1.y = bp[NCH + 16];

            acc0 = __builtin_amdgcn_wmma_f32_16x16x4_f32(
                false, a, false, b0, (short)0, acc0, false, false);
            acc1 = __builtin_amdgcn_wmma_f32_16x16x4_f32(
                false, a, false, b1, (short)0, acc1, false, false);
        }

        // D layout: VGPR r -> pos = mrow+r, channel = nbase(+16)+laneM
        float* vout = &votesLDS[(size_t)cin * MTILE * NCH];
        for (int r = 0; r < 8; ++r) {
            vout[(mrow + r) * NCH + nbase + laneM]      = acc0[r];
            vout[(mrow + r) * NCH + nbase + 16 + laneM] = acc1[r];
        }
    }
    __syncthreads();   // votes tile complete in LDS

    // ---------------- Phase B: dynamic routing -----------------------------
    const int co = t >> 4;
    const int ao = t & 15;
    const float bias_t = bias[t];

    for (int pos = 0; pos < MTILE; ++pos) {
        float v[CIN];
        for (int ci = 0; ci < CIN; ++ci)
            v[ci] = votesLDS[(ci * MTILE + pos) * NCH + t];

        __syncthreads();                     // prior route/logits reads done
        if (t < 128) logits[t >> 4][t & 15] = 0.0f;
        __syncthreads();

        float act = 0.0f;
        for (int it = 0; it < 3; ++it) {
            // softmax over co for each ci (threads 0..127, redundant row scan)
            if (t < 128) {
                const int ci = t >> 4, c = t & 15;
                float mx = logits[ci][0];
                for (int j = 1; j < 16; ++j) mx = fmaxf(mx, logits[ci][j]);
                float s = 0.0f;
                for (int j = 0; j < 16; ++j) s += __expf(logits[ci][j] - mx);
                route[ci][c] = __expf(logits[ci][c] - mx) / s;
            }
            __syncthreads();

            float pre = bias_t;
            for (int ci = 0; ci < CIN; ++ci) pre += route[ci][co] * v[ci];

            // squash: butterfly-reduce pre^2 over the 16 ao lanes (in-wave)
            float s = pre * pre;
            s += __shfl_xor(s, 1);
            s += __shfl_xor(s, 2);
            s += __shfl_xor(s, 4);
            s += __shfl_xor(s, 8);
            const float factor = sqrtf(s) / (1.0f + s);   // ||v||/(1+||v||^2)
            act = pre * factor;

            if (it < 2) {   // agreement -> logits (last iteration unused)
                for (int ci = 0; ci < CIN; ++ci) {
                    float a = v[ci] * act;
                    a += __shfl_xor(a, 1);
                    a += __shfl_xor(a, 2);
                    a += __shfl_xor(a, 4);
                    a += __shfl_xor(a, 8);
                    if (ao == 0) logits[ci][co] += a;   // one writer per (ci,co)
                }
                __syncthreads();
            }
        }

        out[(size_t)(posBase + pos) * NCH + t] = act;
    }
}

// ---------------------------------------------------------------------------
extern "C" void kernel_launch(void* const* d_in, const int* in_sizes, int n_in,
                              void* d_out, int out_size, void* d_ws, size_t ws_size,
                              hipStream_t stream) {
    const float* x    = (const float*)d_in[0];   // [16,32,32,8,32]
    const float* Wt   = (const float*)d_in[1];   // [5,5,32,256]
    const float* bias = (const float*)d_in[2];   // [1,1,16,16]
    float* out = (float*)d_out;                  // [16,32,32,16,16]

    // 1024 blocks: 16*32*32 spatial positions / 16 per block. Fully fused;
    // no workspace needed (votes lives in LDS).
    caps_fused_wmma<<<1024, 256, 0, stream>>>(x, Wt, bias, out);
}